// STSGCN_Attention_62543313764849
// MI455X (gfx1250) — compile-verified
//
#include <hip/hip_runtime.h>
#include <hip/hip_bf16.h>
#include <math.h>

// Problem constants
#define NB   256      // batch N
#define TIN  10
#define TOUT 25
#define VJ   22
#define CC   3
#define HHD  8        // heads
#define DD   66       // V*C
#define KP   68       // DD padded to multiple of 4 (contraction stride)
#define DHD  528      // D*H
#define TTOT 34       // TIN + TOUT - 1
#define G3   198      // 3*D
#define NHB  2048     // N*H
#define SCALEF 0.12309149097933274f   // 1/sqrt(66)

typedef float v2f __attribute__((ext_vector_type(2)));
typedef float v8f __attribute__((ext_vector_type(8)));

__device__ __forceinline__ float sigm(float x) { return 1.f / (1.f + __expf(-x)); }

// ---------------------------------------------------------------------------
// f32 WMMA GEMM:  C[M,N] = A[M,K](lda) * W[N,K](ldw)^T (+bias[n]) (+res) ; opt 2nd store.
// One wave per 16x16 tile, V_WMMA_F32_16X16X4_F32, branch-free inner loop.
// REQUIREMENTS (guaranteed by caller):
//   - K % 4 == 0; pad columns of A and W are zero (or K exact)
//   - lda, ldw even; A, W 8-byte aligned  -> v2f loads are global_load_b64
//   - out-of-range tile rows are safe to *read* after clamping (only stores guarded)
// Fragment layouts per CDNA5 ISA 7.12.2:
//   A 16x4 : a[j] = A[m0+lid][k0 + 2*half + j]
//   B 4x16 : b[j] = W[n0+lid][k0 + 2*half + j]
//   C 16x16: acc[r] = C[m0 + r + 8*half][n0 + lid]
// ---------------------------------------------------------------------------
__global__ void gemm_wmma_f32(const float* __restrict__ A, int lda,
                              const float* __restrict__ W, int ldw,
                              const float* __restrict__ bias,
                              const float* __restrict__ res, int ldres,
                              float* __restrict__ C, int ldc,
                              float* __restrict__ C2, int ldc2,
                              int M, int N, int K)
{
    const int lane = threadIdx.x & 31;
    const int wave = threadIdx.x >> 5;
    const int tile = blockIdx.x * 4 + wave;
    const int tilesN = (N + 15) >> 4;
    const int tilesM = (M + 15) >> 4;
    const int tm = tile / tilesN;
    const int tn = tile - tm * tilesN;
    if (tm >= tilesM) return;                 // wave-uniform exit: EXEC stays full for WMMA
    const int m0 = tm << 4, n0 = tn << 4;
    const int half = lane >> 4, lid = lane & 15;

    // Clamp rows: OOB rows read valid memory; their products only land in
    // C rows/cols we never store.
    const int ar = min(m0 + lid, M - 1);
    const int br = min(n0 + lid, N - 1);
    const float* __restrict__ Arow = A + (size_t)ar * lda + 2 * half;
    const float* __restrict__ Wrow = W + (size_t)br * ldw + 2 * half;

    v8f acc0 = (v8f){0.f,0.f,0.f,0.f,0.f,0.f,0.f,0.f};
    v8f acc1 = acc0;
    int k0 = 0;
    for (; k0 + 8 <= K; k0 += 8) {
        v2f a0 = *(const v2f*)(Arow + k0);
        v2f b0 = *(const v2f*)(Wrow + k0);
        v2f a1 = *(const v2f*)(Arow + k0 + 4);
        v2f b1 = *(const v2f*)(Wrow + k0 + 4);
        acc0 = __builtin_amdgcn_wmma_f32_16x16x4_f32(false, a0, false, b0,
                                                     (short)0, acc0, false, false);
        acc1 = __builtin_amdgcn_wmma_f32_16x16x4_f32(false, a1, false, b1,
                                                     (short)0, acc1, false, false);
    }
    for (; k0 < K; k0 += 4) {
        v2f a0 = *(const v2f*)(Arow + k0);
        v2f b0 = *(const v2f*)(Wrow + k0);
        acc0 = __builtin_amdgcn_wmma_f32_16x16x4_f32(false, a0, false, b0,
                                                     (short)0, acc0, false, false);
    }
    v8f acc = acc0 + acc1;

    const int n = n0 + lid;
    if (n >= N) return;                        // divergent but after all WMMAs
    const float bn = bias ? bias[n] : 0.f;
#pragma unroll
    for (int r = 0; r < 8; ++r) {
        const int m = m0 + r + 8 * half;
        if (m < M) {
            float v = acc[r] + bn;
            if (res) v += res[(size_t)m * ldres + n];
            C[(size_t)m * ldc + n] = v;
            if (C2) C2[(size_t)m * ldc2 + n] = v;
        }
    }
}

// Pad-copy a (rows x DD) weight matrix to (rows x KP), zeroing pad columns.
__global__ void k_pad_w(const float* __restrict__ in, float* __restrict__ out, int rows)
{
    int i = blockIdx.x * blockDim.x + threadIdx.x;
    if (i >= rows * KP) return;
    int r = i / KP, c = i - r * KP;
    out[i] = (c < DD) ? in[r * DD + c] : 0.f;
}

// ---------------------------------------------------------------------------
// ST-GCN stage kernels (tiny work -> plain VALU)
// ---------------------------------------------------------------------------
__global__ void k_transpose_in(const float* __restrict__ x, float* __restrict__ f)
{
    int i = blockIdx.x * blockDim.x + threadIdx.x;
    const int total = NB * CC * TIN * VJ;
    if (i >= total) return;
    int v = i % VJ, t = (i / VJ) % TIN, c = (i / (VJ * TIN)) % CC, n = i / (VJ * TIN * CC);
    f[i] = x[((size_t)(n * TIN + t) * VJ + v) * CC + c];
}

// g1[n,c,q,v] = sum_t f[n,c,t,v] * Tm[v,t,q]
__global__ void k_st_tmix(const float* __restrict__ f, const float* __restrict__ Tm,
                          float* __restrict__ g1, int Ci)
{
    int i = blockIdx.x * blockDim.x + threadIdx.x;
    const int total = NB * Ci * TIN * VJ;
    if (i >= total) return;
    int v = i % VJ, q = (i / VJ) % TIN, nc = i / (VJ * TIN);
    const float* fp = f + (size_t)nc * TIN * VJ + v;
    const float* tp = Tm + v * TIN * TIN + q;
    float s = 0.f;
#pragma unroll
    for (int t = 0; t < TIN; ++t) s += fp[t * VJ] * tp[t * TIN];
    g1[i] = s;
}

// g2[n,c,t,w] = sum_v g1[n,c,t,v] * A[t,v,w]
__global__ void k_st_amix(const float* __restrict__ g1, const float* __restrict__ Aw,
                          float* __restrict__ g2, int Ci)
{
    int i = blockIdx.x * blockDim.x + threadIdx.x;
    const int total = NB * Ci * TIN * VJ;
    if (i >= total) return;
    int w = i % VJ, t = (i / VJ) % TIN, nc = i / (VJ * TIN);
    const float* gp = g1 + (size_t)nc * TIN * VJ + t * VJ;
    const float* ap = Aw + t * VJ * VJ + w;
    float s = 0.f;
#pragma unroll
    for (int v = 0; v < VJ; ++v) s += gp[v] * ap[v * VJ];
    g2[i] = s;
}

// out[n,o,t,v] = PReLU( sum_c g2*cw + cb + sum_c f*rw + rb )
__global__ void k_st_pw(const float* __restrict__ g2, const float* __restrict__ f,
                        const float* __restrict__ cw, const float* __restrict__ cb,
                        const float* __restrict__ rw, const float* __restrict__ rb,
                        const float* __restrict__ aP,
                        float* __restrict__ out, int Ci, int Co)
{
    int i = blockIdx.x * blockDim.x + threadIdx.x;
    const int total = NB * Co * TIN * VJ;
    if (i >= total) return;
    int tv = i % (TIN * VJ), o = (i / (TIN * VJ)) % Co, n = i / (TIN * VJ * Co);
    const float* g2p = g2 + (size_t)n * Ci * TIN * VJ + tv;
    const float* fp  = f  + (size_t)n * Ci * TIN * VJ + tv;
    float s = cb[o] + rb[o];
    for (int c = 0; c < Ci; ++c)
        s += g2p[c * TIN * VJ] * cw[o * Ci + c] + fp[c * TIN * VJ] * rw[o * Ci + c];
    const float a = aP[0];
    out[i] = (s >= 0.f) ? s : a * s;
}

// xf[n,t, d] (row stride KP) = f[n,c,t,v] with d = v*C + c
__global__ void k_make_xf(const float* __restrict__ f, float* __restrict__ xf)
{
    int i = blockIdx.x * blockDim.x + threadIdx.x;
    const int total = NB * TIN * DD;
    if (i >= total) return;
    int d = i % DD, t = (i / DD) % TIN, n = i / (DD * TIN);
    int v = d / CC, c = d % CC;
    xf[(size_t)(n * TIN + t) * KP + d] = f[((size_t)(n * CC + c) * TIN + t) * VJ + v];
}

// out[((n*H+h)*T + t)*D + d] = lin[(n*T+t)*DH + h*D + d]
__global__ void k_reshape_heads(const float* __restrict__ lin, float* __restrict__ o)
{
    int i = blockIdx.x * blockDim.x + threadIdx.x;
    const int total = NHB * TIN * DD;
    if (i >= total) return;
    int d = i % DD, t = (i / DD) % TIN, bh = i / (DD * TIN);
    int n = bh / HHD, h = bh % HHD;
    o[i] = lin[(size_t)(n * TIN + t) * DHD + h * DD + d];
}

// Kb[b, t, :] = kr[b, t, :]   (t < TIN), likewise Vb
__global__ void k_init_kv(const float* __restrict__ kr, const float* __restrict__ vr,
                          float* __restrict__ Kb, float* __restrict__ Vb)
{
    int i = blockIdx.x * blockDim.x + threadIdx.x;
    const int total = NHB * TIN * DD;
    if (i >= total) return;
    int d = i % DD, t = (i / DD) % TIN, b = i / (DD * TIN);
    size_t dst = ((size_t)b * TTOT + t) * DD + d;
    Kb[dst] = kr[i];
    Vb[dst] = vr[i];
}

// Encoder self-attention: one block per (b, query i), T=10 keys. wv row stride KP.
__global__ void k_enc_att(const float* __restrict__ q, const float* __restrict__ k,
                          const float* __restrict__ v, float* __restrict__ wv)
{
    __shared__ float sl[TIN];
    __shared__ float sinv;
    const int bi = blockIdx.x;
    const int b = bi / TIN, iq = bi % TIN;
    const int t = threadIdx.x;
    const float* qp = q + ((size_t)b * TIN + iq) * DD;
    if (t < TIN) {
        const float* kp = k + ((size_t)b * TIN + t) * DD;
        float s = 0.f;
        for (int d = 0; d < DD; ++d) s += qp[d] * kp[d];
        sl[t] = s * SCALEF;
    }
    __syncthreads();
    if (t == 0) {
        float mx = sl[0];
        for (int j = 1; j < TIN; ++j) mx = fmaxf(mx, sl[j]);
        float sm = 0.f;
        for (int j = 0; j < TIN; ++j) { sl[j] = __expf(sl[j] - mx); sm += sl[j]; }
        sinv = 1.f / sm;
    }
    __syncthreads();
    if (t < DD) {
        float s = 0.f;
        for (int j = 0; j < TIN; ++j) s += sl[j] * v[((size_t)b * TIN + j) * DD + t];
        wv[((size_t)b * TIN + iq) * KP + t] = s * sinv;
    }
}

// Decoder masked attention: one block per b; L = valid key count. ctx row stride KP.
__global__ void k_dec_att(const float* __restrict__ qi, const float* __restrict__ Kb,
                          const float* __restrict__ Vb, float* __restrict__ ctx, int L)
{
    __shared__ float sl[TTOT];
    __shared__ float sinv;
    const int b = blockIdx.x;
    const int t = threadIdx.x;
    const float* qp = qi + (size_t)b * DD;
    if (t < L) {
        const float* kp = Kb + ((size_t)b * TTOT + t) * DD;
        float s = 0.f;
        for (int d = 0; d < DD; ++d) s += qp[d] * kp[d];
        sl[t] = s * SCALEF;
    }
    __syncthreads();
    if (t == 0) {
        float mx = sl[0];
        for (int j = 1; j < L; ++j) mx = fmaxf(mx, sl[j]);
        float sm = 0.f;
        for (int j = 0; j < L; ++j) { sl[j] = __expf(sl[j] - mx); sm += sl[j]; }
        sinv = 1.f / sm;
    }
    __syncthreads();
    if (t < DD) {
        float s = 0.f;
        for (int j = 0; j < L; ++j) s += sl[j] * Vb[((size_t)b * TTOT + j) * DD + t];
        ctx[(size_t)b * KP + t] = s * sinv;
    }
}

// ki/vi are (NHB,66) views of the (NB,528) projection outputs (stride DD within DHD rows)
__global__ void k_scatter_kv(const float* __restrict__ ki, const float* __restrict__ vi,
                             float* __restrict__ Kb, float* __restrict__ Vb, int pos)
{
    int i = blockIdx.x * blockDim.x + threadIdx.x;
    const int total = NHB * DD;
    if (i >= total) return;
    int d = i % DD, b = i / DD;
    size_t dst = ((size_t)b * TTOT + pos) * DD + d;
    Kb[dst] = ki[i];
    Vb[dst] = vi[i];
}

// GRU cell (both directions). gi row for batch b at giX + b*giXs. h row stride KP.
__global__ void k_gru_cell(const float* __restrict__ giF, int giFs, const float* __restrict__ ghF,
                           const float* __restrict__ giB, int giBs, const float* __restrict__ ghB,
                           float* __restrict__ hf, float* __restrict__ hb)
{
    int i = blockIdx.x * blockDim.x + threadIdx.x;
    const int total = NHB * DD;
    if (i >= total) return;
    int d = i % DD, b = i / DD;
    size_t hix = (size_t)b * KP + d;
    {
        const float* gi = giF + (size_t)b * giFs;
        const float* gh = ghF + (size_t)b * G3;
        float r = sigm(gi[d] + gh[d]);
        float z = sigm(gi[DD + d] + gh[DD + d]);
        float nn = tanhf(gi[2 * DD + d] + r * gh[2 * DD + d]);
        hf[hix] = (1.f - z) * nn + z * hf[hix];
    }
    {
        const float* gi = giB + (size_t)b * giBs;
        const float* gh = ghB + (size_t)b * G3;
        float r = sigm(gi[d] + gh[d]);
        float z = sigm(gi[DD + d] + gh[DD + d]);
        float nn = tanhf(gi[2 * DD + d] + r * gh[2 * DD + d]);
        hb[hix] = (1.f - z) * nn + z * hb[hix];
    }
}

// havg compact (N, DH): havg[b*DD+d] = 0.5*(hf[b*KP+d] + hb[b*KP+d])
__global__ void k_avg_h(const float* __restrict__ hf, const float* __restrict__ hb,
                        float* __restrict__ havg)
{
    int i = blockIdx.x * blockDim.x + threadIdx.x;
    if (i >= NHB * DD) return;
    int d = i % DD, b = i / DD;
    size_t hix = (size_t)b * KP + d;
    havg[i] = 0.5f * (hf[hix] + hb[hix]);
}

// ---------------------------------------------------------------------------
static inline int cdiv(int a, int b) { return (a + b - 1) / b; }

extern "C" void kernel_launch(void* const* d_in, const int* in_sizes, int n_in,
                              void* d_out, int out_size, void* d_ws, size_t ws_size,
                              hipStream_t stream)
{
    (void)in_sizes; (void)n_in; (void)out_size; (void)ws_size;
    const float* X = (const float*)d_in[0];
    const float *Aw[4], *Tm[4], *cwp[4], *cbp[4], *rwp[4], *rbp[4], *alp[4];
    for (int l = 0; l < 4; ++l) {
        int base = 1 + l * 7;
        Aw[l]  = (const float*)d_in[base + 0];
        Tm[l]  = (const float*)d_in[base + 1];
        cwp[l] = (const float*)d_in[base + 2];
        cbp[l] = (const float*)d_in[base + 3];
        rwp[l] = (const float*)d_in[base + 4];
        rbp[l] = (const float*)d_in[base + 5];
        alp[l] = (const float*)d_in[base + 6];
    }
    const float* Wq = (const float*)d_in[29];
    const float* Wk = (const float*)d_in[30];
    const float* Wv = (const float*)d_in[31];
    const float* gwf_ih = (const float*)d_in[32];
    const float* gwf_hh = (const float*)d_in[33];
    const float* gbf_ih = (const float*)d_in[34];
    const float* gbf_hh = (const float*)d_in[35];
    const float* gwb_ih = (const float*)d_in[36];
    const float* gwb_hh = (const float*)d_in[37];
    const float* gbb_ih = (const float*)d_in[38];
    const float* gbb_hh = (const float*)d_in[39];
    const float* Lw = (const float*)d_in[40];
    const float* Lb = (const float*)d_in[41];
    float* dout = (float*)d_out;

    // Workspace bump allocator (floats), 16B-aligned chunks
    float* wsf = (float*)d_ws;
    size_t off = 0;
    auto alloc = [&](size_t nelem) { float* p = wsf + off; off += (nelem + 3) & ~(size_t)3; return p; };
    const size_t FMAX = (size_t)NB * 64 * TIN * VJ;
    float* fA   = alloc(FMAX);
    float* fB   = alloc(FMAX);
    float* g1   = alloc(FMAX);
    float* g2   = alloc(FMAX);
    float* xf   = alloc((size_t)NB * TIN * KP);       // padded contraction stride
    float* lin  = alloc((size_t)NB * TIN * DHD);
    float* qr   = alloc((size_t)NHB * TIN * DD);
    float* kr   = alloc((size_t)NHB * TIN * DD);
    float* vr   = alloc((size_t)NHB * TIN * DD);
    float* wv   = alloc((size_t)NHB * TIN * KP);      // padded
    float* gif  = alloc((size_t)NHB * TIN * G3);
    float* gib  = alloc((size_t)NHB * TIN * G3);
    float* ghf  = alloc((size_t)NHB * G3);
    float* ghb  = alloc((size_t)NHB * G3);
    float* hf   = alloc((size_t)NHB * KP);            // padded
    float* hb   = alloc((size_t)NHB * KP);            // padded
    float* havg = alloc((size_t)NHB * DD);            // compact (N, DH)
    float* Kb   = alloc((size_t)NHB * TTOT * DD);
    float* Vb   = alloc((size_t)NHB * TTOT * DD);
    float* yhat = alloc((size_t)NB * KP);             // padded
    float* dqi  = alloc((size_t)NB * DHD);            // == (NHB, DD) reinterpreted
    float* dki  = alloc((size_t)NB * DHD);
    float* dvi  = alloc((size_t)NB * DHD);
    float* ctx  = alloc((size_t)NHB * KP);            // padded
    // padded weights (rows x KP)
    float* WqP   = alloc((size_t)DHD * KP);
    float* WkP   = alloc((size_t)DHD * KP);
    float* WvP   = alloc((size_t)DHD * KP);
    float* gfihP = alloc((size_t)G3 * KP);
    float* gfhhP = alloc((size_t)G3 * KP);
    float* gbihP = alloc((size_t)G3 * KP);
    float* gbhhP = alloc((size_t)G3 * KP);

    auto gemm = [&](const float* A, int lda, const float* W, int ldw, const float* bias,
                    const float* res, int ldres, float* C, int ldc,
                    float* C2, int ldc2, int M, int Nn, int K) {
        int tiles = cdiv(M, 16) * cdiv(Nn, 16);
        gemm_wmma_f32<<<cdiv(tiles, 4), 128, 0, stream>>>(
            A, lda, W, ldw, bias, res, ldres, C, ldc, C2, ldc2, M, Nn, K);
    };
    auto ew = [&](int total) { return cdiv(total, 256); };

    // Zero padded activation buffers (pad columns must be 0 for the GEMMs).
    hipMemsetAsync(xf,   0, (size_t)NB * TIN * KP * sizeof(float), stream);
    hipMemsetAsync(wv,   0, (size_t)NHB * TIN * KP * sizeof(float), stream);
    hipMemsetAsync(hf,   0, (size_t)NHB * KP * sizeof(float), stream);
    hipMemsetAsync(hb,   0, (size_t)NHB * KP * sizeof(float), stream);
    hipMemsetAsync(ctx,  0, (size_t)NHB * KP * sizeof(float), stream);
    hipMemsetAsync(yhat, 0, (size_t)NB * KP * sizeof(float), stream);

    // Pad weights to K=68
    k_pad_w<<<ew(DHD * KP), 256, 0, stream>>>(Wq, WqP, DHD);
    k_pad_w<<<ew(DHD * KP), 256, 0, stream>>>(Wk, WkP, DHD);
    k_pad_w<<<ew(DHD * KP), 256, 0, stream>>>(Wv, WvP, DHD);
    k_pad_w<<<ew(G3 * KP), 256, 0, stream>>>(gwf_ih, gfihP, G3);
    k_pad_w<<<ew(G3 * KP), 256, 0, stream>>>(gwf_hh, gfhhP, G3);
    k_pad_w<<<ew(G3 * KP), 256, 0, stream>>>(gwb_ih, gbihP, G3);
    k_pad_w<<<ew(G3 * KP), 256, 0, stream>>>(gwb_hh, gbhhP, G3);

    // ---------------- ST-GCN stack ----------------
    k_transpose_in<<<ew(NB * CC * TIN * VJ), 256, 0, stream>>>(X, fA);
    const int ci[4] = {3, 64, 32, 64};
    const int co[4] = {64, 32, 64, 3};
    float* fin = fA;
    float* fout = fB;
    for (int l = 0; l < 4; ++l) {
        int tot = NB * ci[l] * TIN * VJ;
        k_st_tmix<<<ew(tot), 256, 0, stream>>>(fin, Tm[l], g1, ci[l]);
        k_st_amix<<<ew(tot), 256, 0, stream>>>(g1, Aw[l], g2, ci[l]);
        int toto = NB * co[l] * TIN * VJ;
        k_st_pw<<<ew(toto), 256, 0, stream>>>(g2, fin, cwp[l], cbp[l], rwp[l], rbp[l],
                                              alp[l], fout, ci[l], co[l]);
        float* tmp = fin; fin = fout; fout = tmp;
    }
    k_make_xf<<<ew(NB * TIN * DD), 256, 0, stream>>>(fin, xf);

    // ---------------- QKV projections + head reshape ----------------
    gemm(xf, KP, WqP, KP, nullptr, nullptr, 0, lin, DHD, nullptr, 0, NB * TIN, DHD, KP);
    k_reshape_heads<<<ew(NHB * TIN * DD), 256, 0, stream>>>(lin, qr);
    gemm(xf, KP, WkP, KP, nullptr, nullptr, 0, lin, DHD, nullptr, 0, NB * TIN, DHD, KP);
    k_reshape_heads<<<ew(NHB * TIN * DD), 256, 0, stream>>>(lin, kr);
    gemm(xf, KP, WvP, KP, nullptr, nullptr, 0, lin, DHD, nullptr, 0, NB * TIN, DHD, KP);
    k_reshape_heads<<<ew(NHB * TIN * DD), 256, 0, stream>>>(lin, vr);

    k_init_kv<<<ew(NHB * TIN * DD), 256, 0, stream>>>(kr, vr, Kb, Vb);

    // ---------------- Encoder self-attention ----------------
    k_enc_att<<<NHB * TIN, 96, 0, stream>>>(qr, kr, vr, wv);

    // ---------------- Bidirectional GRU encoder ----------------
    // gate-input GEMMs for all 10 steps at once (M = 20480)
    gemm(wv, KP, gfihP, KP, gbf_ih, nullptr, 0, gif, G3, nullptr, 0, NHB * TIN, G3, KP);
    gemm(wv, KP, gbihP, KP, gbb_ih, nullptr, 0, gib, G3, nullptr, 0, NHB * TIN, G3, KP);
    for (int t = 0; t < TIN; ++t) {
        gemm(hf, KP, gfhhP, KP, gbf_hh, nullptr, 0, ghf, G3, nullptr, 0, NHB, G3, KP);
        gemm(hb, KP, gbhhP, KP, gbb_hh, nullptr, 0, ghb, G3, nullptr, 0, NHB, G3, KP);
        k_gru_cell<<<ew(NHB * DD), 256, 0, stream>>>(
            gif + (size_t)t * G3, TIN * G3, ghf,
            gib + (size_t)(TIN - 1 - t) * G3, TIN * G3, ghb, hf, hb);
    }

    // ---------------- yhat0 ----------------
    k_avg_h<<<ew(NHB * DD), 256, 0, stream>>>(hf, hb, havg);
    gemm(havg, DHD, Lw, DHD, Lb, xf + (size_t)(TIN - 1) * KP, TIN * KP,
         yhat, KP, dout, TOUT * DD, NB, DD, DHD);

    // ---------------- Autoregressive decoder ----------------
    for (int i = 0; i < TOUT - 1; ++i) {
        // projections of yhat (256x66 pad 68): output (256x528) == per-head (2048x66)
        gemm(yhat, KP, WqP, KP, nullptr, nullptr, 0, dqi, DHD, nullptr, 0, NB, DHD, KP);
        gemm(yhat, KP, WkP, KP, nullptr, nullptr, 0, dki, DHD, nullptr, 0, NB, DHD, KP);
        gemm(yhat, KP, WvP, KP, nullptr, nullptr, 0, dvi, DHD, nullptr, 0, NB, DHD, KP);
        k_dec_att<<<NHB, 128, 0, stream>>>(dqi, Kb, Vb, ctx, TIN + i);
        k_scatter_kv<<<ew(NHB * DD), 256, 0, stream>>>(dki, dvi, Kb, Vb, TIN + i);
        // GRU cells on ctx (batch 2048)
        gemm(ctx, KP, gfihP, KP, gbf_ih, nullptr, 0, gif, G3, nullptr, 0, NHB, G3, KP);
        gemm(hf,  KP, gfhhP, KP, gbf_hh, nullptr, 0, ghf, G3, nullptr, 0, NHB, G3, KP);
        gemm(ctx, KP, gbihP, KP, gbb_ih, nullptr, 0, gib, G3, nullptr, 0, NHB, G3, KP);
        gemm(hb,  KP, gbhhP, KP, gbb_hh, nullptr, 0, ghb, G3, nullptr, 0, NHB, G3, KP);
        k_gru_cell<<<ew(NHB * DD), 256, 0, stream>>>(gif, G3, ghf, gib, G3, ghb, hf, hb);
        // yhat += Linear(0.5*(hf+hb)); also store into output slot i+1
        k_avg_h<<<ew(NHB * DD), 256, 0, stream>>>(hf, hb, havg);
        gemm(havg, DHD, Lw, DHD, Lb, yhat, KP,
             yhat, KP, dout + (size_t)(i + 1) * DD, TOUT * DD, NB, DD, DHD);
    }
}